// BipartiteCamBEVGAT_60567628808234
// MI455X (gfx1250) — compile-verified
//
#include <hip/hip_runtime.h>
#include <hip/hip_bf16.h>

typedef __attribute__((ext_vector_type(2))) float v2f;
typedef __attribute__((ext_vector_type(8))) float v8f;

// ---------------------------------------------------------------------------
// utility
// ---------------------------------------------------------------------------
__device__ __forceinline__ void atomicMaxF(float* addr, float val) {
  // monotonic mapping trick: works for mixed signs, init with -inf
  if (val >= 0.f) atomicMax((int*)addr, __float_as_int(val));
  else            atomicMin((unsigned int*)addr, __float_as_uint(val));
}

__global__ void fill_f32(float* __restrict__ p, float v, long long n) {
  long long i = blockIdx.x * (long long)blockDim.x + threadIdx.x;
  if (i < n) p[i] = v;
}

// ---------------------------------------------------------------------------
// self-loop edge_attr = mean of incoming edge_attr (fill_value='mean')
// ---------------------------------------------------------------------------
__global__ void deg_attr_sum(const int* __restrict__ dst, const float* __restrict__ eattr,
                             float* __restrict__ deg, float* __restrict__ asum, int E) {
  int e = blockIdx.x * blockDim.x + threadIdx.x;
  if (e >= E) return;
  int d = dst[e];
  atomicAdd(&deg[d], 1.0f);
#pragma unroll
  for (int j = 0; j < 8; ++j) atomicAdd(&asum[d * 8 + j], eattr[(long long)e * 8 + j]);
}

__global__ void self_attr_fin(float* __restrict__ asum, const float* __restrict__ deg, int Nn) {
  int i = blockIdx.x * blockDim.x + threadIdx.x;
  if (i >= Nn * 8) return;
  float dg = deg[i >> 3];
  dg = dg > 1.f ? dg : 1.f;
  asum[i] = asum[i] / dg;
}

// ---------------------------------------------------------------------------
// Dual projection via fp32 WMMA (16x16x4), one wave per 16x16 output tile:
//   Yl[N,M] = X[N,K] @ Wl[K,M] + bl      Yr[N,M] = X[N,K] @ Wr[K,M] + br
// A fragment loaded once per k-step, feeds both accumulators.
// A frag (16x4 f32, 2 VGPRs): lane<16 -> K pair {k0,k0+1}, lane>=16 -> {k0+2,k0+3},
//                             row = lane&15.  B frag symmetric with col = lane&15.
// C/D (8 VGPRs): row = tile + vgpr + 8*(lane>=16), col = tile + (lane&15).
// ---------------------------------------------------------------------------
__global__ void linear2_wmma(const float* __restrict__ X,
                             const float* __restrict__ Wl, const float* __restrict__ bl,
                             const float* __restrict__ Wr, const float* __restrict__ br,
                             float* __restrict__ Yl, float* __restrict__ Yr,
                             int Nrows, int K, int M) {
  int lane = threadIdx.x & 31;
  int wave = (int)((blockIdx.x * (long long)blockDim.x + threadIdx.x) >> 5);
  int mTiles = (M + 15) >> 4;
  int nTiles = (Nrows + 15) >> 4;
  if (wave >= mTiles * nTiles) return;               // wave-uniform exit
  int rowTile = wave / mTiles;
  int colTile = wave - rowTile * mTiles;
  int l16 = lane & 15;
  int row = rowTile * 16 + l16;
  if (row >= Nrows) row = Nrows - 1;                 // clamp (N is 16-aligned here)
  int col = colTile * 16 + l16;
  bool colOk = col < M;
  int colC = colOk ? col : 0;
  int kOff = (lane >> 4) << 1;                       // 0 or 2
  const float* xrow = X + (long long)row * K;
  v8f accL = {};
  v8f accR = {};
#pragma unroll 4
  for (int k0 = 0; k0 < K; k0 += 4) {
    int ka = k0 + kOff;
    v2f a, b, c;
    a.x = xrow[ka];
    a.y = xrow[ka + 1];
    float bl0 = Wl[(long long)ka * M + colC];
    float bl1 = Wl[(long long)(ka + 1) * M + colC];
    float br0 = Wr[(long long)ka * M + colC];
    float br1 = Wr[(long long)(ka + 1) * M + colC];
    b.x = colOk ? bl0 : 0.f;
    b.y = colOk ? bl1 : 0.f;
    c.x = colOk ? br0 : 0.f;
    c.y = colOk ? br1 : 0.f;
    accL = __builtin_amdgcn_wmma_f32_16x16x4_f32(false, a, false, b, (short)0, accL,
                                                 false, false);
    accR = __builtin_amdgcn_wmma_f32_16x16x4_f32(false, a, false, c, (short)0, accR,
                                                 false, false);
  }
  if (colOk) {
    float bbl = bl[col];
    float bbr = br[col];
    int rBase = rowTile * 16 + ((lane >> 4) << 3);
#pragma unroll
    for (int v = 0; v < 8; ++v) {
      int r = rBase + v;
      if (r < Nrows) {
        Yl[(long long)r * M + col] = accL[v] + bbl;
        Yr[(long long)r * M + col] = accR[v] + bbr;
      }
    }
  }
}

// ---------------------------------------------------------------------------
// per (edge, head): logit = att . leaky_relu(xl[s] + xr[d] + ea@We); seg-max
// edges [0,E) are real, [E, E+N) are self-loops with mean edge_attr
// ---------------------------------------------------------------------------
template <int H, int C>
__global__ void edge_logit(const int* __restrict__ src, const int* __restrict__ dst,
                           const float* __restrict__ eattr, const float* __restrict__ sattr,
                           const float* __restrict__ xl, const float* __restrict__ xr,
                           const float* __restrict__ We, const float* __restrict__ att,
                           float* __restrict__ logit, float* __restrict__ mx,
                           int E, int Nn) {
  const int HC = H * C;
  int tid = blockIdx.x * blockDim.x + threadIdx.x;
  int total = (E + Nn) * H;
  if (tid >= total) return;
  int e = tid / H;
  int h = tid - e * H;
  int s, d;
  const float* ea;
  if (e < E) { s = src[e]; d = dst[e]; ea = eattr + (long long)e * 8; }
  else       { s = d = e - E;          ea = sattr + (long long)(e - E) * 8; }
  float eav[8];
#pragma unroll
  for (int j = 0; j < 8; ++j) eav[j] = ea[j];
  const float* xls  = xl  + (long long)s * HC + h * C;
  const float* xrd  = xr  + (long long)d * HC + h * C;
  const float* Wec  = We  + h * C;
  const float* attc = att + h * C;
  float acc = 0.f;
#pragma unroll
  for (int c = 0; c < C; ++c) {
    float ep = 0.f;
#pragma unroll
    for (int j = 0; j < 8; ++j) ep = fmaf(eav[j], Wec[j * HC + c], ep);
    float m = xls[c] + xrd[c] + ep;
    m = m > 0.f ? m : 0.2f * m;                      // leaky_relu(0.2)
    acc = fmaf(attc[c], m, acc);
  }
  logit[tid] = acc;
  atomicMaxF(&mx[d * H + h], acc);
}

// ---------------------------------------------------------------------------
// fused exp + denom accumulation + UNNORMALIZED scatter:
//   agg[d] += exp(logit - mx[d]) * xl[s];  denom[d] += exp(...)
// the 1/denom normalization is linear, so it is deferred to the finalize pass
// (one divide per node-channel instead of per edge-head) — saves a full edge
// sweep vs separate exp and scatter kernels.
// ---------------------------------------------------------------------------
template <int H, int C>
__global__ void edge_exp_scatter(const int* __restrict__ src, const int* __restrict__ dst,
                                 const float* __restrict__ logit, const float* __restrict__ mx,
                                 float* __restrict__ denom, const float* __restrict__ xl,
                                 float* __restrict__ agg, int E, int Nn) {
  const int HC = H * C;
  int tid = blockIdx.x * blockDim.x + threadIdx.x;
  int total = (E + Nn) * H;
  if (tid >= total) return;
  int e = tid / H;
  int h = tid - e * H;
  int s, d;
  if (e < E) { s = src[e]; d = dst[e]; }
  else       { s = d = e - E; }
  float a = __expf(logit[tid] - mx[d * H + h]);
  atomicAdd(&denom[d * H + h], a);
  const float* xls = xl + (long long)s * HC + h * C;
  float* o = agg + (long long)d * HC + h * C;
#pragma unroll
  for (int c = 0; c < C; ++c) atomicAdd(&o[c], a * xls[c]);
}

// finalize layers 0/1: y = ELU(agg/denom + bias)
__global__ void norm_bias_elu(float* __restrict__ y, const float* __restrict__ denom,
                              const float* __restrict__ bias, int Nn, int H, int C) {
  const int HC = H * C;
  int i = blockIdx.x * blockDim.x + threadIdx.x;
  if (i >= Nn * HC) return;
  int n = i / HC;
  int rem = i - n * HC;
  int h = rem / C;
  float v = y[i] / denom[n * H + h] + bias[rem];
  y[i] = v > 0.f ? v : (__expf(v) - 1.f);            // ELU(alpha=1)
}

// finalize layer 2: out = mean_h(agg/denom) + bias
__global__ void norm_head_mean(const float* __restrict__ agg, const float* __restrict__ denom,
                               const float* __restrict__ bias, float* __restrict__ out,
                               int Nn, int H, int C) {
  int i = blockIdx.x * blockDim.x + threadIdx.x;
  if (i >= Nn * C) return;
  int n = i / C;
  int c = i - n * C;
  float s = 0.f;
  for (int h = 0; h < H; ++h)
    s += agg[(long long)n * H * C + h * C + c] / denom[n * H + h];
  out[i] = s * (1.0f / H) + bias[c];
}

// ---------------------------------------------------------------------------
// launch
// ---------------------------------------------------------------------------
extern "C" void kernel_launch(void* const* d_in, const int* in_sizes, int n_in,
                              void* d_out, int out_size, void* d_ws, size_t ws_size,
                              hipStream_t stream) {
  if (n_in < 24) return;
  const float* x     = (const float*)d_in[0];
  const int*   eidx  = (const int*)d_in[1];
  const float* eattr = (const float*)d_in[2];
  const int E  = in_sizes[1] / 2;
  const int Nn = in_sizes[0] / 256;
  const int* src = eidx;
  const int* dst = eidx + E;

  // params dict order per layer: Wl, bl, Wr, br, We, att, bias
  const float* P[3][7];
  for (int l = 0; l < 3; ++l)
    for (int j = 0; j < 7; ++j)
      P[l][j] = (const float*)d_in[3 + l * 7 + j];

  float* ws = (float*)d_ws;
  long long off = 0;
  auto carve = [&](long long n) { float* p = ws + off; off += n; return p; };
  float* deg   = carve(Nn);
  float* sattr = carve((long long)Nn * 8);
  float* bufA  = carve((long long)Nn * 128);
  float* bufB  = carve((long long)Nn * 128);
  float* xl    = carve((long long)Nn * 128);
  float* xr    = carve((long long)Nn * 128);
  float* logit = carve((long long)(E + Nn) * 4);
  float* mx    = carve((long long)Nn * 4);
  float* denom = carve((long long)Nn * 4);
  (void)ws_size;

  const int BT = 256;
  auto nb = [&](long long n) { return (unsigned)((n + BT - 1) / BT); };

  // self-loop edge_attr = segment_mean(edge_attr, dst)
  fill_f32<<<nb(Nn), BT, 0, stream>>>(deg, 0.f, Nn);
  fill_f32<<<nb((long long)Nn * 8), BT, 0, stream>>>(sattr, 0.f, (long long)Nn * 8);
  deg_attr_sum<<<nb(E), BT, 0, stream>>>(dst, eattr, deg, sattr, E);
  self_attr_fin<<<nb((long long)Nn * 8), BT, 0, stream>>>(sattr, deg, Nn);

  const int H = 4;
  const int total = (E + Nn) * H;
  const float NEG_INF = -__builtin_inff();

  const float* hin = x;
  for (int l = 0; l < 3; ++l) {
    const int fi = (l == 0) ? 256 : 128;
    const int C  = (l == 2) ? 6 : 32;
    const int HC = H * C;
    float* agg = (l == 0) ? bufA : (l == 1 ? bufB : bufA);   // ping-pong

    // fused dual projection via fp32 WMMA (A fragments read once)
    int mT = (HC + 15) >> 4, nT = (Nn + 15) >> 4;
    long long waves = (long long)mT * nT;
    unsigned gblk = (unsigned)((waves * 32 + BT - 1) / BT);
    linear2_wmma<<<gblk, BT, 0, stream>>>(hin, P[l][0], P[l][1], P[l][2], P[l][3],
                                          xl, xr, Nn, fi, HC);

    // segment-softmax state
    fill_f32<<<nb((long long)Nn * H), BT, 0, stream>>>(mx, NEG_INF, (long long)Nn * H);
    fill_f32<<<nb((long long)Nn * H), BT, 0, stream>>>(denom, 0.f, (long long)Nn * H);
    fill_f32<<<nb((long long)Nn * HC), BT, 0, stream>>>(agg, 0.f, (long long)Nn * HC);

    if (C == 32) {
      edge_logit<4, 32><<<nb(total), BT, 0, stream>>>(src, dst, eattr, sattr, xl, xr,
                                                      P[l][4], P[l][5], logit, mx, E, Nn);
      edge_exp_scatter<4, 32><<<nb(total), BT, 0, stream>>>(src, dst, logit, mx, denom,
                                                            xl, agg, E, Nn);
    } else {
      edge_logit<4, 6><<<nb(total), BT, 0, stream>>>(src, dst, eattr, sattr, xl, xr,
                                                     P[l][4], P[l][5], logit, mx, E, Nn);
      edge_exp_scatter<4, 6><<<nb(total), BT, 0, stream>>>(src, dst, logit, mx, denom,
                                                           xl, agg, E, Nn);
    }

    if (l < 2) {
      norm_bias_elu<<<nb((long long)Nn * HC), BT, 0, stream>>>(agg, denom, P[l][6],
                                                               Nn, H, C);
      hin = agg;
    } else {
      norm_head_mean<<<nb((long long)Nn * 6), BT, 0, stream>>>(agg, denom, P[l][6],
                                                               (float*)d_out, Nn, H, 6);
    }
  }
}